// DifferentiableShockProximity_86638080295378
// MI455X (gfx1250) — compile-verified
//
#include <hip/hip_runtime.h>
#include <cstdint>

// Problem constants from the reference: state is (64, 1, 2048) fp32, dx = 1/2048.
#define BATCH 64
#define NX    2048
#define TPB   256            // 8 wave32 per block
#define PER   (NX / TPB)     // 8 cells per thread
#define NWAVE (TPB / 32)     // 8 waves
#define SIGMA 0.05f
#define NEG_SENT (-(1 << 30))
#define POS_SENT (1 << 30)

__global__ __launch_bounds__(TPB) void shock_proximity_kernel(
    const float* __restrict__ state,
    const float* __restrict__ dxp,
    float* __restrict__ out)
{
    __shared__ float s_row[NX + 4];   // row + sentinel pad (branch-free mask)
    __shared__ int   s_wmax[NWAVE];   // per-wave max shock index (or sentinel)
    __shared__ int   s_wmin[NWAVE];   // per-wave min shock index (or sentinel)

    const int t    = threadIdx.x;
    const int lane = t & 31;
    const int w    = t >> 5;
    const int b    = blockIdx.x;
    const int j0   = t * PER;
    const float* row = state + (size_t)b * NX;
    const float  dx  = dxp[0];        // uniform scalar load, hoisted early

    // ---- CDNA5 async copy: global -> LDS, 32 B per lane (2x B128), GVS mode.
    // LDS address from the VDST vgpr plus offset; global address = SGPR base + VGPR offset + offset
    {
        unsigned lds_addr = (unsigned)(uintptr_t)(void*)&s_row[j0]; // low 32b of flat LDS ptr == LDS byte offset
        unsigned goff     = (unsigned)j0 * 4u;
        asm volatile(
            "global_load_async_to_lds_b128 %0, %1, %2 offset:0\n\t"
            "global_load_async_to_lds_b128 %0, %1, %2 offset:16\n\t"
            "s_wait_asynccnt 0x0"
            :
            : "v"(lds_addr), "v"(goff), "s"(row)
            : "memory");
    }
    if (t == 0) s_row[NX] = -1e30f;   // sentinel: interface j = NX-1 never a shock
    __syncthreads();                  // barrier #1

    // ---- branch-free mask over interfaces j in [j0, j0+PER): shock <=> s[j] < s[j+1]
    float v[PER + 1];
    {
        const float4* p = (const float4*)&s_row[j0];   // 32B-aligned
        float4 a = p[0], c = p[1];
        v[0] = a.x; v[1] = a.y; v[2] = a.z; v[3] = a.w;
        v[4] = c.x; v[5] = c.y; v[6] = c.z; v[7] = c.w;
        v[8] = s_row[j0 + PER];
    }
    int gloc[PER], hloc[PER];
    int tmax = NEG_SENT, tmin = POS_SENT;
#pragma unroll
    for (int k = 0; k < PER; ++k) {
        bool m  = v[k] < v[k + 1];
        int  j  = j0 + k;
        gloc[k] = m ? j : NEG_SENT;
        hloc[k] = m ? j : POS_SENT;
        tmax = max(tmax, gloc[k]);
        tmin = min(tmin, hloc[k]);
    }

    // ---- wave32 nearest-neighbor via ballot (tmax/tmin are monotone in lane):
    // nearest-left source = highest ballot lane below me; nearest-right = lowest above me.
    const unsigned B  = (unsigned)__ballot(tmax >= 0);     // lanes whose chunk has a shock
    const unsigned pm = B & ((1u << lane) - 1u);           // earlier lanes
    const unsigned sm = B & (0xFFFFFFFEu << lane);         // later lanes
    const int srcHi  = 31 - __clz((int)pm);                // pm==0 -> -1, selected out below
    const int srcLo  = __ffs((int)sm) - 1;                 // sm==0 -> -1, selected out below
    const int srcTop = 31 - __clz((int)B);
    const int srcBot = __ffs((int)B) - 1;
    const int vHi  = __builtin_amdgcn_ds_bpermute(srcHi  << 2, tmax);
    const int vLo  = __builtin_amdgcn_ds_bpermute(srcLo  << 2, tmin);
    const int vTop = __builtin_amdgcn_ds_bpermute(srcTop << 2, tmax);
    const int vBot = __builtin_amdgcn_ds_bpermute(srcBot << 2, tmin);
    const int exclMax = pm ? vHi : NEG_SENT;   // max shock j over lanes < lane
    const int exclMin = sm ? vLo : POS_SENT;   // min shock j over lanes > lane

    // ---- cross-wave combine: 8 aggregates through LDS, one barrier
    if (lane == 0) {
        s_wmax[w] = B ? vTop : NEG_SENT;       // wave-wide max shock index
        s_wmin[w] = B ? vBot : POS_SENT;       // wave-wide min shock index
    }
    __syncthreads();                           // barrier #2

    int wavePref = NEG_SENT, waveSuf = POS_SENT, allMax = NEG_SENT;
#pragma unroll
    for (int q = 0; q < NWAVE; ++q) {
        int mx = s_wmax[q];
        int mn = s_wmin[q];
        allMax = max(allMax, mx);
        if (q < w) wavePref = max(wavePref, mx);
        if (q > w) waveSuf  = min(waveSuf, mn);
    }
    const int prefMax = max(wavePref, exclMax);   // max shock j over interfaces j < j0
    const int sufMin  = min(waveSuf,  exclMin);   // min shock j over interfaces j >= j0+PER
    const float anyF  = (allMax >= 0) ? 1.0f : 0.0f;

    // ---- per-cell nearest distance, exp, store
    int rsuf[PER + 1];
    rsuf[PER] = sufMin;
#pragma unroll
    for (int k = PER - 1; k >= 0; --k) rsuf[k] = min(hloc[k], rsuf[k + 1]);

    float res[PER];
    int runL = prefMax;
#pragma unroll
    for (int k = 0; k < PER; ++k) {
        const int i    = j0 + k;
        const int Lidx = runL;      // nearest shock j <= i-1 (or sentinel)
        const int Ridx = rsuf[k];   // nearest shock j >= i   (or sentinel)
        // exact fp32: dist = |i - j - 0.5| * dx  (dx = 2^-11, all steps exact)
        const float dl   = (float)(i - Lidx) - 0.5f;
        const float dr   = (float)(Ridx - i) + 0.5f;
        const float dmin = fminf(dl, dr) * dx;
        res[k] = __expf(-dmin / SIGMA) * anyF;
        runL = max(runL, gloc[k]);
    }
    // 32B contiguous per thread -> two global_store_b128
    float4* op = (float4*)(out + (size_t)b * NX + j0);
    op[0] = make_float4(res[0], res[1], res[2], res[3]);
    op[1] = make_float4(res[4], res[5], res[6], res[7]);
}

extern "C" void kernel_launch(void* const* d_in, const int* in_sizes, int n_in,
                              void* d_out, int out_size, void* d_ws, size_t ws_size,
                              hipStream_t stream) {
    (void)in_sizes; (void)n_in; (void)d_ws; (void)ws_size; (void)out_size;
    const float* state = (const float*)d_in[0];
    const float* dx    = (const float*)d_in[1];
    float*       out   = (float*)d_out;
    shock_proximity_kernel<<<dim3(BATCH), dim3(TPB), 0, stream>>>(state, dx, out);
}